// TripletLoss_88570815578691
// MI455X (gfx1250) — compile-verified
//
#include <hip/hip_runtime.h>
#include <hip/hip_bf16.h>

typedef __attribute__((ext_vector_type(2))) float v2f;
typedef __attribute__((ext_vector_type(8))) float v8f;

#define B_N 4096
#define D_K 512

// ---------------- JAX Threefry-2x32 (exact) ----------------
__device__ __forceinline__ unsigned rotl32(unsigned x, unsigned d) {
    return (x << d) | (x >> (32u - d));
}
__device__ __forceinline__ void tf4(unsigned& x0, unsigned& x1,
                                    unsigned r0, unsigned r1, unsigned r2, unsigned r3) {
    x0 += x1; x1 = rotl32(x1, r0); x1 ^= x0;
    x0 += x1; x1 = rotl32(x1, r1); x1 ^= x0;
    x0 += x1; x1 = rotl32(x1, r2); x1 ^= x0;
    x0 += x1; x1 = rotl32(x1, r3); x1 ^= x0;
}
__device__ __forceinline__ void threefry2x32(unsigned k0, unsigned k1,
                                             unsigned& x0, unsigned& x1) {
    unsigned ks2 = k0 ^ k1 ^ 0x1BD11BDAu;
    x0 += k0; x1 += k1;
    tf4(x0, x1, 13, 15, 26, 6);  x0 += k1;  x1 += ks2 + 1u;
    tf4(x0, x1, 17, 29, 16, 24); x0 += ks2; x1 += k0  + 2u;
    tf4(x0, x1, 13, 15, 26, 6);  x0 += k0;  x1 += k1  + 3u;
    tf4(x0, x1, 17, 29, 16, 24); x0 += k1;  x1 += ks2 + 4u;
    tf4(x0, x1, 13, 15, 26, 6);  x0 += ks2; x1 += k0  + 5u;
}
// uniform[f] of a (B_N,B_N) jax.random.uniform draw, flat index f
__device__ __forceinline__ float jax_uniform01(unsigned k0, unsigned k1,
                                               unsigned f, unsigned half) {
    bool lo = f < half;
    unsigned x0 = lo ? f : (f - half);
    unsigned x1 = lo ? (f + half) : f;
    threefry2x32(k0, k1, x0, x1);
    unsigned bits = lo ? x0 : x1;
    return __uint_as_float((bits >> 9) | 0x3F800000u) - 1.0f;
}

// ---------------- K1: squared norms (one wave per row) + zero accumulators ----
__global__ __launch_bounds__(256) void tl_sqnorm_kernel(const float* __restrict__ E,
                                                        float* __restrict__ sq,
                                                        float* sumAcc, unsigned* cntAcc) {
    if (blockIdx.x == 0 && threadIdx.x == 0) { *sumAcc = 0.0f; *cntAcc = 0u; }
    const int wave = threadIdx.x >> 5;
    const int lane = threadIdx.x & 31;
    const int row  = blockIdx.x * 8 + wave;
    const float* R = E + (size_t)row * D_K;
    float acc = 0.0f;
#pragma unroll
    for (int t = 0; t < D_K / 32; ++t) {
        float v = R[lane + 32 * t];
        acc = fmaf(v, v, acc);
    }
    for (int off = 16; off > 0; off >>= 1) acc += __shfl_xor(acc, off, 32);
    if (lane == 0) sq[row] = acc;
}

// ---------------- K2: masked argmax of JAX uniforms (one block per row) -------
__global__ __launch_bounds__(256) void tl_argmax_kernel(const int* __restrict__ labels,
                                                        int* __restrict__ idxp,
                                                        int* __restrict__ idxn) {
    __shared__ float sv[256];
    __shared__ int   si[256];
    const int row = blockIdx.x;
    const int tid = threadIdx.x;
    const int labi = labels[row];

    // kp, kn = jax.random.split(jax.random.key(42)); key words = (0, 42)
    unsigned a0 = 0u, a1 = 2u; threefry2x32(0u, 42u, a0, a1); // y0/y1 of ctr (0,2)
    unsigned b0 = 1u, b1 = 3u; threefry2x32(0u, 42u, b0, b1); // y0/y1 of ctr (1,3)
    const unsigned kp0 = a0, kp1 = b0;  // first split key
    const unsigned kn0 = a1, kn1 = b1;  // second split key

    const unsigned half = ((unsigned)B_N * (unsigned)B_N) / 2u;

    float bpv = -1.0f, bnv = -1.0f;
    int   bpi = 0,     bni = 0;
    for (int j = tid; j < B_N; j += 256) {
        int labj = labels[j];
        unsigned f = (unsigned)row * (unsigned)B_N + (unsigned)j;
        float up = jax_uniform01(kp0, kp1, f, half);
        float un = jax_uniform01(kn0, kn1, f, half);
        bool mp = (labj == labi) && (j != row);
        bool mn = (labj != labi);
        float vp = mp ? up : -1.0f;
        float vn = mn ? un : -1.0f;
        if (vp > bpv) { bpv = vp; bpi = j; }   // strict > keeps lowest j per thread
        if (vn > bnv) { bnv = vn; bni = j; }
    }
    // reduce positives (tie-break: larger value, then smaller index)
    sv[tid] = bpv; si[tid] = bpi; __syncthreads();
    for (int s = 128; s > 0; s >>= 1) {
        if (tid < s) {
            float v2 = sv[tid + s]; int i2 = si[tid + s];
            if (v2 > sv[tid] || (v2 == sv[tid] && i2 < si[tid])) { sv[tid] = v2; si[tid] = i2; }
        }
        __syncthreads();
    }
    if (tid == 0) idxp[row] = (sv[0] >= 0.0f) ? si[0] : -1;
    __syncthreads();
    // reduce negatives
    sv[tid] = bnv; si[tid] = bni; __syncthreads();
    for (int s = 128; s > 0; s >>= 1) {
        if (tid < s) {
            float v2 = sv[tid + s]; int i2 = si[tid + s];
            if (v2 > sv[tid] || (v2 == sv[tid] && i2 < si[tid])) { sv[tid] = v2; si[tid] = i2; }
        }
        __syncthreads();
    }
    if (tid == 0) idxn[row] = (sv[0] >= 0.0f) ? si[0] : -1;
}

// ---------------- K3: WMMA gather-dot + triplet loss (1 wave = 16 rows) -------
__global__ __launch_bounds__(32) void tl_wmma_kernel(const float* __restrict__ E,
                                                     const float* __restrict__ sq,
                                                     const int* __restrict__ idxp,
                                                     const int* __restrict__ idxn,
                                                     float* sumAcc, unsigned* cntAcc) {
    __shared__ float lp[8][32];
    __shared__ float ln_[8][32];
    const int lane = threadIdx.x;
    const int m    = lane & 15;   // row within tile (A: M index, B: N index)
    const int half = lane >> 4;   // 0 -> K={0,1}, 1 -> K={2,3}
    const int g    = blockIdx.x;
    const int rowA = g * 16 + m;

    int pi_raw = idxp[rowA];
    int ni_raw = idxn[rowA];
    int pi = pi_raw < 0 ? 0 : pi_raw;
    int ni = ni_raw < 0 ? 0 : ni_raw;

    const float* Arow = E + (size_t)rowA * D_K + 2 * half;
    const float* Prow = E + (size_t)pi   * D_K + 2 * half;
    const float* Nrow = E + (size_t)ni   * D_K + 2 * half;

    v8f cp = {0.f, 0.f, 0.f, 0.f, 0.f, 0.f, 0.f, 0.f};
    v8f cn = {0.f, 0.f, 0.f, 0.f, 0.f, 0.f, 0.f, 0.f};
#pragma unroll 4
    for (int k0 = 0; k0 < D_K; k0 += 4) {
        v2f a  = *(const v2f*)(Arow + k0);
        v2f bp = *(const v2f*)(Prow + k0);
        v2f bn = *(const v2f*)(Nrow + k0);
        cp = __builtin_amdgcn_wmma_f32_16x16x4_f32(false, a, false, bp, (short)0, cp, false, false);
        cn = __builtin_amdgcn_wmma_f32_16x16x4_f32(false, a, false, bn, (short)0, cn, false, false);
    }
#pragma unroll
    for (int r = 0; r < 8; ++r) { lp[r][lane] = cp[r]; ln_[r][lane] = cn[r]; }
    __syncthreads();

    float t = 0.0f;
    unsigned c = 0u;
    if (lane < 16) {
        // diagonal (m,m) of C: M<8 -> VGPR m, lane m ; M>=8 -> VGPR m-8, lane m+16
        int rr = (lane < 8) ? lane : lane - 8;
        int ll = (lane < 8) ? lane : lane + 16;
        float dotp = lp[rr][ll];
        float dotn = ln_[rr][ll];
        float si = sq[rowA];
        float dp = (pi_raw >= 0) ? fmaxf(0.0f, si + sq[pi] - 2.0f * dotp) : 0.0f;
        float dn = (ni_raw >= 0) ? fmaxf(0.0f, si + sq[ni] - 2.0f * dotn) : 0.0f;
        t = fmaxf(0.0f, dp - dn + 1.0f);   // MARGIN = 1.0
        c = (t > 1e-16f) ? 1u : 0u;
    }
    for (int off = 16; off > 0; off >>= 1) {
        t += __shfl_down(t, off, 32);
        c += __shfl_down(c, off, 32);
    }
    if (lane == 0) { atomicAdd(sumAcc, t); atomicAdd(cntAcc, c); }
}

// ---------------- K4: finalize ------------------------------------------------
__global__ void tl_final_kernel(const float* sumAcc, const unsigned* cntAcc, float* out) {
    if (threadIdx.x == 0 && blockIdx.x == 0) {
        out[0] = (*cntAcc == 0u) ? 0.0f : (*sumAcc) * (1.0f / (float)B_N);
    }
}

extern "C" void kernel_launch(void* const* d_in, const int* in_sizes, int n_in,
                              void* d_out, int out_size, void* d_ws, size_t ws_size,
                              hipStream_t stream) {
    const float* E      = (const float*)d_in[0];  // [4096, 512] fp32
    const int*   labels = (const int*)d_in[1];    // [4096] int32
    float*       out    = (float*)d_out;          // scalar fp32

    // workspace layout (floats)
    float*    ws     = (float*)d_ws;
    float*    sumAcc = ws;                               // [1]
    unsigned* cntAcc = (unsigned*)(ws + 1);              // [1]
    float*    sq     = ws + 16;                          // [4096]
    int*      idxp   = (int*)(ws + 16 + B_N);            // [4096]
    int*      idxn   = (int*)(ws + 16 + 2 * B_N);        // [4096]

    tl_sqnorm_kernel<<<B_N / 8, 256, 0, stream>>>(E, sq, sumAcc, cntAcc);
    tl_argmax_kernel<<<B_N, 256, 0, stream>>>(labels, idxp, idxn);
    tl_wmma_kernel<<<B_N / 16, 32, 0, stream>>>(E, sq, idxp, idxn, sumAcc, cntAcc);
    tl_final_kernel<<<1, 1, 0, stream>>>(sumAcc, cntAcc, out);
}